// Rotation_64802466562550
// MI455X (gfx1250) — compile-verified
//
#include <hip/hip_runtime.h>
#include <hip/hip_bf16.h>

typedef float v2f __attribute__((ext_vector_type(2)));
typedef float v8f __attribute__((ext_vector_type(8)));

#define DDIM 128
#define CDIM 32
#define RREF 8
#define WSTRIDE 130   // pad 128->130: keeps 8B alignment for ds_load_b64, avoids bank conflicts

__global__ __launch_bounds__(256, 1)
void rotation_householder_kernel(const float* __restrict__ src,
                                 const float* __restrict__ data,
                                 const float* __restrict__ W,
                                 const float* __restrict__ a,
                                 const float* __restrict__ c,
                                 const float* __restrict__ d,
                                 float* __restrict__ out)
{
    // Double-buffered W[r] tiles: async-load r+1 while WMMA consumes r.
    __shared__ float ldsW[2][DDIM * WSTRIDE];   // 2 x 66,560 B = 130 KB (< 320 KB/WGP)

    const int tid  = threadIdx.x;
    const int wave = tid >> 5;
    const int lane = tid & 31;
    const int half = lane >> 4;     // 0: lanes 0-15, 1: lanes 16-31
    const int ln   = lane & 15;

    const int rb = blockIdx.x * 128 + wave * 16;   // wave's first row in [B*C]

    // Issue async copy of W[r] (row-major, padded) into LDS buffer r&1.
    // CDNA5 GLOBAL_LOAD_ASYNC_TO_LDS_B64, tracked by ASYNCcnt (no VGPR data).
    auto issueW = [&](int r) {
        const float* wr = W + (size_t)r * DDIM * DDIM;
        float* dst = &ldsW[r & 1][0];
        for (int i = tid; i < DDIM * DDIM / 2; i += 256) {
            const int row  = i >> 6;
            const int col2 = (i & 63) * 2;
            unsigned lds_off = (unsigned)(uintptr_t)(dst + row * WSTRIDE + col2);
            unsigned long long ga =
                (unsigned long long)(uintptr_t)(wr + row * DDIM + col2);
            asm volatile("global_load_async_to_lds_b64 %0, %1, off"
                         :: "v"(lds_off), "v"(ga) : "memory");
        }
    };

    issueW(0);   // overlaps with the A-fragment / data loads below

    // ---- A fragments for the wave's 16x128 source tile, cached in registers.
    // 16x4 fp32 A layout: lanes 0-15 row=ln K={k4,k4+1}; lanes 16-31 row=ln K={k4+2,k4+3}
    v2f afrag[32];
    {
        const float* srow = src + (size_t)(rb + ln) * DDIM + 2 * half;
        #pragma unroll
        for (int ks = 0; ks < 32; ++ks)
            afrag[ks] = *(const v2f*)(srow + ks * 4);
    }

    // ---- data rows in C/D-matrix register layout: xr[t][j] = data[rb + j + 8*half][t*16 + ln]
    v8f xr[8];
    #pragma unroll
    for (int t = 0; t < 8; ++t) {
        #pragma unroll
        for (int j = 0; j < 8; ++j)
            xr[t][j] = data[(size_t)(rb + j + 8 * half) * DDIM + t * 16 + ln];
    }

    const v8f vzero = {};
    const int cbase = rb & (CDIM - 1);

    for (int r = 0; r < RREF; ++r) {
        // Wait for this thread's async chunks of W[r]; barrier makes all chunks visible.
        asm volatile("s_wait_asynccnt 0x0" ::: "memory");
        __syncthreads();
        if (r + 1 < RREF)
            issueW(r + 1);                 // prefetch next W into the other buffer

        const float* wbuf = ldsW[r & 1];

        // ---- GEMM: v_tile(16x128) = A(16x128) x W[r]^T via v_wmma_f32_16x16x4_f32
        v8f acc[8];
        #pragma unroll
        for (int t = 0; t < 8; ++t) acc[t] = vzero;

        #pragma unroll
        for (int ks = 0; ks < 32; ++ks) {
            const int kk = ks * 4 + 2 * half;
            #pragma unroll
            for (int t = 0; t < 8; ++t) {
                // B(4x16) fragment: lane -> column n = t*16+ln, K pair {kk,kk+1}
                v2f b = *(const v2f*)&wbuf[(t * 16 + ln) * WSTRIDE + kk];
                acc[t] = __builtin_amdgcn_wmma_f32_16x16x4_f32(
                    false, afrag[ks], false, b, (short)0, acc[t], false, false);
            }
        }

        // ---- DynamicTanh (hardware V_TANH_F32) + L2-normalize + Householder, in registers.
        // Row m = j + 8*half lives in VGPR j of each n-tile, spread over one 16-lane half.
        #pragma unroll
        for (int j = 0; j < 8; ++j) {
            const int cidx = (cbase + j + 8 * half) & (CDIM - 1);
            const float* arow = a + ((size_t)r * CDIM + cidx) * DDIM;
            const float* crow = c + ((size_t)r * CDIM + cidx) * DDIM;
            const float* drow = d + ((size_t)r * CDIM + cidx) * DDIM;

            float ss = 0.0f;
            #pragma unroll
            for (int t = 0; t < 8; ++t) {
                const int o = t * 16 + ln;
                const float av = arow[o] * acc[t][j];
                float th;
                // TRANS op: keep 2 V_NOPs inside so the result/source hazard window
                // (1 independent op rule) is satisfied regardless of scheduling.
                asm volatile("v_tanh_f32 %0, %1\n\tv_nop\n\tv_nop"
                             : "=v"(th) : "v"(av));
                const float v = th * crow[o] + drow[o];
                acc[t][j] = v;
                ss += v * v;
            }
            // reduce over the 16 lanes of this half (wave32; masks 1,2,4,8 stay in-half)
            ss += __shfl_xor(ss, 1, 32);
            ss += __shfl_xor(ss, 2, 32);
            ss += __shfl_xor(ss, 4, 32);
            ss += __shfl_xor(ss, 8, 32);
            const float inv = 1.0f / fmaxf(sqrtf(ss), 1e-12f);

            float dotp = 0.0f;
            #pragma unroll
            for (int t = 0; t < 8; ++t) {
                const float v = acc[t][j] * inv;
                acc[t][j] = v;
                dotp += v * xr[t][j];
            }
            dotp += __shfl_xor(dotp, 1, 32);
            dotp += __shfl_xor(dotp, 2, 32);
            dotp += __shfl_xor(dotp, 4, 32);
            dotp += __shfl_xor(dotp, 8, 32);
            const float s2 = 2.0f * dotp;

            #pragma unroll
            for (int t = 0; t < 8; ++t)
                xr[t][j] -= s2 * acc[t][j];
        }
    }

    // ---- store reflected rows
    #pragma unroll
    for (int t = 0; t < 8; ++t) {
        #pragma unroll
        for (int j = 0; j < 8; ++j)
            out[(size_t)(rb + j + 8 * half) * DDIM + t * 16 + ln] = xr[t][j];
    }
}

extern "C" void kernel_launch(void* const* d_in, const int* in_sizes, int n_in,
                              void* d_out, int out_size, void* d_ws, size_t ws_size,
                              hipStream_t stream) {
    const float* src  = (const float*)d_in[0];
    const float* data = (const float*)d_in[1];
    const float* W    = (const float*)d_in[2];
    const float* a    = (const float*)d_in[3];
    const float* c    = (const float*)d_in[4];
    const float* d    = (const float*)d_in[5];

    const int M = in_sizes[0] / DDIM;       // B*C = 131072 rows
    const int blocks = M / 128;             // 128 rows per 256-thread block (8 waves x 16 rows)
    rotation_householder_kernel<<<blocks, 256, 0, stream>>>(
        src, data, W, a, c, d, (float*)d_out);
}